// BBoxDecoder_17506286698708
// MI455X (gfx1250) — compile-verified
//
#include <hip/hip_runtime.h>
#include <math.h>

// ---------------------------------------------------------------------------
// Problem constants (match reference)
// ---------------------------------------------------------------------------
#define BB      32          // batch
#define NPTS    512         // timestamps per sample
#define LAT     4096        // latent
#define H1DIM   512
#define H2DIM   1024
#define HID     256         // CoordinateNet hidden
#define PTOT    198916      // flat param count per sample

// per-sample offsets inside flat param vector
#define OFF_WIN   0
#define OFF_BIN   256
#define OFF_W0    512
#define OFF_B0    66048
#define OFF_W1    66304
#define OFF_B1    131840
#define OFF_W2    132096
#define OFF_B2    197632
#define OFF_WOUT  197888
#define OFF_BOUT  198912

// workspace layout (float element offsets)
#define WS_H1     0u
#define WS_H2     (WS_H1 + BB*H1DIM)                 // 16384
#define WS_FLAT   (WS_H2 + BB*H2DIM)                 // 49152
#define WS_XA     (WS_FLAT + (size_t)BB*PTOT)        // 6414464
#define WS_XB     (WS_XA + (size_t)BB*NPTS*HID)      // 10608768

typedef float v2f __attribute__((ext_vector_type(2)));
typedef float v8f __attribute__((ext_vector_type(8)));
typedef int   v4i __attribute__((__vector_size__(4 * sizeof(int))));

#define V8F_ZERO {0.f,0.f,0.f,0.f,0.f,0.f,0.f,0.f}

#define GLOBAL_AS __attribute__((address_space(1)))
#define LDS_AS    __attribute__((address_space(3)))

// ---------------------------------------------------------------------------
// f32 WMMA step: D = A(16x4) x B(4x16) + C, wave32.
// A frag: lane%16 = M row, (lane/16)*2 + {0,1} = K.  B frag symmetric.
// ---------------------------------------------------------------------------
__device__ __forceinline__ v8f wmma4(v2f a, v2f b, v8f c) {
  return __builtin_amdgcn_wmma_f32_16x16x4_f32(
      /*neg_a=*/false, a, /*neg_b=*/false, b,
      /*c_mod=*/(short)0, c, /*reuse_a=*/false, /*reuse_b=*/false);
}

// Async copy of 16B global -> LDS (GLOBAL_LOAD_ASYNC_TO_LDS_B128, ASYNCcnt).
// Builtin params are (global v4i*, local v4i*, imm offset, imm cpol).
__device__ __forceinline__ void async_load_b128(const float* gsrc, float* ldst) {
  __builtin_amdgcn_global_load_async_to_lds_b128(
      (GLOBAL_AS v4i*)gsrc, (LDS_AS v4i*)ldst, /*offset=*/0, /*cpol=*/0);
}

// Generic 16x16 tile of  Out = X (MxK, ld=ldx) @ W^T (W is NxK, ld=ldw)
__device__ __forceinline__ v8f gemm_tile_f32(const float* __restrict__ X, int ldx,
                                             const float* __restrict__ W, int ldw,
                                             int m0, int n0, int K, v8f acc) {
  const int lane = threadIdx.x & 31;
  const int hi   = lane >> 4;      // 0 / 1
  const int l16  = lane & 15;
  const float* xr = X + (size_t)(m0 + l16) * ldx;
  const float* wr = W + (size_t)(n0 + l16) * ldw;
  #pragma unroll 4
  for (int k = 0; k < K; k += 4) {
    const int kk = k + hi * 2;
    v2f a; a.x = xr[kk]; a.y = xr[kk + 1];
    v2f b; b.x = wr[kk]; b.y = wr[kk + 1];
    acc = wmma4(a, b, acc);
  }
  return acc;
}

// C/D layout: VGPR r, lanes 0-15 -> M=m0+r, lanes 16-31 -> M=m0+8+r; N = n0 + lane%16
__device__ __forceinline__ void store_tile_bias_act(float* __restrict__ Out, int ldo,
                                                    const float* __restrict__ bias,
                                                    int m0, int n0, v8f c, bool relu_act) {
  const int lane = threadIdx.x & 31;
  const int hi   = lane >> 4;
  const int col  = n0 + (lane & 15);
  const float bv = bias[col];
  #pragma unroll
  for (int r = 0; r < 8; ++r) {
    const int row = m0 + r + hi * 8;
    float v = c[r] + bv;
    if (relu_act) v = fmaxf(v, 0.f);
    Out[(size_t)row * ldo + col] = v;
  }
}

// ---------------------------------------------------------------------------
// K1/K2: small hypernet GEMMs, one wave per 16x16 tile. Out = relu(X @ W^T + b)
// ---------------------------------------------------------------------------
__global__ void hyper_gemm_relu(const float* __restrict__ X,
                                const float* __restrict__ W,
                                const float* __restrict__ bias,
                                float* __restrict__ Out,
                                int M, int N, int K) {
  const int wid = (blockIdx.x * blockDim.x + threadIdx.x) >> 5;
  const int ntiles = N >> 4;
  const int mtiles = M >> 4;
  if (wid >= ntiles * mtiles) return;          // uniform per wave
  const int mt = wid / ntiles, nt = wid % ntiles;
  v8f c = V8F_ZERO;
  c = gemm_tile_f32(X, K, W, K, mt * 16, nt * 16, K, c);
  store_tile_bias_act(Out, N, bias, mt * 16, nt * 16, c, true);
}

// ---------------------------------------------------------------------------
// K3: flat = h2 (32x1024) @ W3^T + b3 ;  W3 is PTOT x 1024 (815 MB stream).
// Memory-bound leg of the pipeline: double-buffered LDS, next K-chunk streams
// in via GLOBAL_LOAD_ASYNC_TO_LDS_B128 while WMMA consumes the current one.
// s_wait_asynccnt 8 retires only the previous chunk's 8 per-wave async ops,
// leaving the prefetch in flight. Pad LDS rows to 68 f32 to spread banks.
// ---------------------------------------------------------------------------
#define K3_KC 64
#define K3_LDS_LD 68
#define K3_NCHUNK (H2DIM / K3_KC)   // 16

__global__ void gemm3_kernel(const float* __restrict__ X,      // 32 x 1024
                             const float* __restrict__ W,      // PTOT x 1024
                             const float* __restrict__ bias,   // PTOT
                             float* __restrict__ Out) {        // 32 x PTOT
  __shared__ alignas(16) float lds[2][128 * K3_LDS_LD];
  const int t    = threadIdx.x;
  const int lane = t & 31;
  const int wv   = t >> 5;                 // wave 0..7
  const int hi   = lane >> 4;
  const int l16  = lane & 15;
  const int blockRow0 = blockIdx.x * 128;  // first W3 row handled by block
  const int n0 = blockRow0 + wv * 16;      // this wave's output-column tile

  v8f c0 = V8F_ZERO;   // M rows 0..15
  v8f c1 = V8F_ZERO;   // M rows 16..31

  const float* xr0 = X + (size_t)l16 * H2DIM;
  const float* xr1 = X + (size_t)(16 + l16) * H2DIM;

  // Issue one 128-row x 64-f32 chunk: 8 b128 async ops per thread (coalesced:
  // consecutive threads hit consecutive float4 slots along a W3 row).
  auto issue_chunk = [&](int kc, int buf) {
    #pragma unroll
    for (int i = 0; i < 8; ++i) {
      const int s   = t + i * 256;     // float4 slot
      const int row = s >> 4;          // 16 float4 per row
      const int c4  = s & 15;
      int grow = blockRow0 + row;
      if (grow >= PTOT) grow = PTOT - 1;   // clamp: those rows feed cols >= PTOT,
                                           // which the store guard discards
      async_load_b128(W + (size_t)grow * H2DIM + kc + c4 * 4,
                      &lds[buf][row * K3_LDS_LD + c4 * 4]);
    }
  };

  issue_chunk(0, 0);
  for (int i = 0; i < K3_NCHUNK; ++i) {
    const int kc = i * K3_KC;
    if (i + 1 < K3_NCHUNK) {
      issue_chunk(kc + K3_KC, (i + 1) & 1);            // prefetch next chunk
      asm volatile("s_wait_asynccnt 0x8" ::: "memory"); // chunk i landed
    } else {
      asm volatile("s_wait_asynccnt 0x0" ::: "memory");
    }
    __syncthreads();                                    // all waves' chunk i in LDS

    const float* wr = &lds[i & 1][(wv * 16 + l16) * K3_LDS_LD];
    #pragma unroll
    for (int k = 0; k < K3_KC; k += 4) {
      const int kk = k + hi * 2;
      const int gk = kc + kk;
      v2f a0; a0.x = xr0[gk]; a0.y = xr0[gk + 1];
      v2f a1; a1.x = xr1[gk]; a1.y = xr1[gk + 1];
      v2f b;  b.x  = wr[kk];  b.y  = wr[kk + 1];
      c0 = wmma4(a0, b, c0);
      c1 = wmma4(a1, b, c1);
    }
    __syncthreads();   // done reading buf[i&1] before it is refilled at i+2
  }

  const int col = n0 + l16;
  if (col < PTOT) {
    const float bv = bias[col];
    #pragma unroll
    for (int r = 0; r < 8; ++r) {
      const int row = r + hi * 8;
      Out[(size_t)row * PTOT + col]        = c0[r] + bv;
      Out[(size_t)(row + 16) * PTOT + col] = c1[r] + bv;
    }
  }
}

// ---------------------------------------------------------------------------
// K4a: CoordinateNet input layer (elementwise): x0[b][n][h] = relu(t*w_in[h]+b_in[h])
// ---------------------------------------------------------------------------
__global__ void coord_input(const float* __restrict__ ts,     // B x N x 1
                            const float* __restrict__ flat,   // B x PTOT
                            float* __restrict__ X0) {         // B x N x HID
  const int idx = blockIdx.x * blockDim.x + threadIdx.x;      // B*N*HID threads
  const int h = idx & (HID - 1);
  const int n = (idx >> 8) & (NPTS - 1);
  const int b = idx >> 17;
  const float* p = flat + (size_t)b * PTOT;
  const float tv = ts[b * NPTS + n];
  const float v  = tv * p[OFF_WIN + h] + p[OFF_BIN + h];
  X0[idx] = fmaxf(v, 0.f);
}

// ---------------------------------------------------------------------------
// K4b: per-sample hidden GEMM: Xout = relu(Xin (512x256) @ Wb^T + bb), wave/tile.
// Params live in L2 (25 MB flat), so direct loads are acceptable here.
// ---------------------------------------------------------------------------
__global__ void coord_hidden(const float* __restrict__ Xin,   // B x N x HID
                             const float* __restrict__ flat,  // B x PTOT
                             int offW, int offB,
                             float* __restrict__ Xout) {
  const int wid = (blockIdx.x * blockDim.x + threadIdx.x) >> 5;
  // tiles per sample: (512/16) x (256/16) = 32 x 16 = 512
  const int s   = wid >> 9;
  const int rem = wid & 511;
  const int mt  = rem >> 4;
  const int nt  = rem & 15;
  const float* X = Xin  + (size_t)s * NPTS * HID;
  const float* W = flat + (size_t)s * PTOT + offW;
  const float* bias = flat + (size_t)s * PTOT + offB;
  float* Out = Xout + (size_t)s * NPTS * HID;
  v8f c = V8F_ZERO;
  c = gemm_tile_f32(X, HID, W, HID, mt * 16, nt * 16, HID, c);
  store_tile_bias_act(Out, HID, bias, mt * 16, nt * 16, c, true);
}

// ---------------------------------------------------------------------------
// K4c: output layer, N=4: plain dot products + sigmoid (tiny: 17 M MAC)
// ---------------------------------------------------------------------------
__global__ void coord_output(const float* __restrict__ X3,    // B x N x HID
                             const float* __restrict__ flat,  // B x PTOT
                             float* __restrict__ Out) {       // B x N x 4
  const int tid = blockIdx.x * blockDim.x + threadIdx.x;      // B*N*4 threads
  const int o = tid & 3;
  const int n = (tid >> 2) & (NPTS - 1);
  const int b = tid >> 11;
  const float* p  = flat + (size_t)b * PTOT;
  const float* wo = p + OFF_WOUT + o * HID;
  const float* xr = X3 + ((size_t)b * NPTS + n) * HID;
  float acc = p[OFF_BOUT + o];
  #pragma unroll 8
  for (int k = 0; k < HID; ++k) acc += xr[k] * wo[k];
  Out[tid] = 1.f / (1.f + __expf(-acc));
}

// ---------------------------------------------------------------------------
// launch
// ---------------------------------------------------------------------------
extern "C" void kernel_launch(void* const* d_in, const int* in_sizes, int n_in,
                              void* d_out, int out_size, void* d_ws, size_t ws_size,
                              hipStream_t stream) {
  const float* z  = (const float*)d_in[0];
  const float* ts = (const float*)d_in[1];
  const float* W1 = (const float*)d_in[2];
  const float* b1 = (const float*)d_in[3];
  const float* W2 = (const float*)d_in[4];
  const float* b2 = (const float*)d_in[5];
  const float* W3 = (const float*)d_in[6];
  const float* b3 = (const float*)d_in[7];
  float* out = (float*)d_out;

  float* ws   = (float*)d_ws;
  float* h1   = ws + WS_H1;
  float* h2   = ws + WS_H2;
  float* flat = ws + WS_FLAT;
  float* xA   = ws + WS_XA;
  float* xB   = ws + WS_XB;

  // HyperNet
  hyper_gemm_relu<<<8, 256, 0, stream>>>(z, W1, b1, h1, BB, H1DIM, LAT);     // 64 tiles
  hyper_gemm_relu<<<16, 256, 0, stream>>>(h1, W2, b2, h2, BB, H2DIM, H1DIM); // 128 tiles
  const int ntiles3  = (PTOT + 15) / 16;             // 12433
  const int nblocks3 = (ntiles3 + 7) / 8;            // 1555
  gemm3_kernel<<<nblocks3, 256, 0, stream>>>(h2, W3, b3, flat);

  // CoordinateNet
  coord_input<<<(BB * NPTS * HID) / 256, 256, 0, stream>>>(ts, flat, xA);
  const int hidden_blocks = (BB * 512) / 8;          // 16384 waves / 8 per block
  coord_hidden<<<hidden_blocks, 256, 0, stream>>>(xA, flat, OFF_W0, OFF_B0, xB);
  coord_hidden<<<hidden_blocks, 256, 0, stream>>>(xB, flat, OFF_W1, OFF_B1, xA);
  coord_hidden<<<hidden_blocks, 256, 0, stream>>>(xA, flat, OFF_W2, OFF_B2, xB);
  coord_output<<<(BB * NPTS * 4) / 256, 256, 0, stream>>>(xB, flat, out);
}